// GraphNetwork_37160057045415
// MI455X (gfx1250) — compile-verified
//
#include <hip/hip_runtime.h>

#define N_NODES 100000
#define N_EDGES 1250000
#define NFEAT   64
#define EDGE_DIM 13
#define EPS 1e-5f

typedef __attribute__((ext_vector_type(2))) float        v2f;
typedef __attribute__((ext_vector_type(8))) float        v8f;
typedef __attribute__((ext_vector_type(4))) unsigned int v4u;
typedef __attribute__((ext_vector_type(8))) int          v8i_;
typedef __attribute__((ext_vector_type(4))) int          v4i_;

__device__ __forceinline__ v8f wmma_f32(v2f a, v2f b, v8f c) {
  // D = A(16x4) * B(4x16) + C(16x16), full fp32 (v_wmma_f32_16x16x4_f32)
  return __builtin_amdgcn_wmma_f32_16x16x4_f32(false, a, false, b, (short)0, c,
                                               false, false);
}

__device__ __forceinline__ float sigmoidf_(float x) {
  return 1.0f / (1.0f + __expf(-x));
}

// -------------------------------------------------------------------------
// TDM: DMA a [rows x cols] fp32 row-major matrix from global into LDS.
// D# per CDNA5 ISA (2-D tile == whole tensor,
// groups 2/3 zero). Issue once per wave; tracked with TENSORcnt.
// -------------------------------------------------------------------------
__device__ __forceinline__ void tdm_load_2d(const float* gsrc, float* ldst,
                                            unsigned rows, unsigned cols) {
  const unsigned long long ga = (unsigned long long)(size_t)gsrc;
  const unsigned la = (unsigned)(size_t)ldst;  // LDS aperture: offset in addr[31:0]
  v4u g0;
  g0[0] = 1u;                                   // count=1, is_restore=0, gather=0
  g0[1] = la;                                   // lds_addr (bytes)
  g0[2] = (unsigned)ga;                         // global_addr[31:0]
  g0[3] = (unsigned)((ga >> 32) & 0x01FFFFFFu)  // global_addr[56:32]
          | 0x80000000u;                        // type=2 ("image")
  v8i_ g1;
  g1[0] = (int)(2u << 16);                      // data_size=2 -> 4 bytes
  g1[1] = (int)((cols & 0xFFFFu) << 16);        // tensor_dim0[15:0]
  g1[2] = (int)((cols >> 16) | ((rows & 0xFFFFu) << 16));  // td0[31:16] | td1[15:0]
  g1[3] = (int)((rows >> 16) | (cols << 16));   // td1[31:16] | tile_dim0
  g1[4] = (int)(rows & 0xFFFFu);                // tile_dim1 (tile_dim2=0)
  g1[5] = (int)cols;                            // tensor_dim0_stride[31:0]
  g1[6] = 0;                                    // stride0[47:32] | stride1[15:0]
  g1[7] = 0;                                    // stride1[47:16]
  const v4i_ z4 = {0, 0, 0, 0};
  const v8i_ z8 = {0, 0, 0, 0, 0, 0, 0, 0};
  __builtin_amdgcn_tensor_load_to_lds(g0, g1, z4, z4, z8, 0);
}

// -------------------------------------------------------------------------
// fnet: per-edge MLP  13 -> 64 -> 64 -> 64 (relu, relu, linear+bias)
// One wave per 16-edge tile. Layer1 via VALU directly into A-layout,
// layers 2/3 via chained f32 WMMA; W2/W3 staged in LDS by the TDM.
// K-step-outer / tile-inner ordering -> 4 independent WMMA chains.
// -------------------------------------------------------------------------
__global__ void fnet_kernel(const float* __restrict__ ef,
                            const float* __restrict__ W1, const float* __restrict__ b1,
                            const float* __restrict__ W2, const float* __restrict__ b2,
                            const float* __restrict__ W3, const float* __restrict__ b3,
                            float* __restrict__ w_out) {
  __shared__ float conv[8][16][68];      // per-wave C->A layout staging (padded)
  extern __shared__ float dynlds[];      // [0..4095]=W2, [4096..8191]=W3
  float* lW2 = dynlds;
  float* lW3 = dynlds + 4096;

  const int wave = threadIdx.x >> 5;
  const int lane = threadIdx.x & 31;
  const int col  = lane & 15;
  const int half = lane >> 4;
  const int kb   = half * 2;
  const long long e0 = ((long long)blockIdx.x * 8 + wave) * 16;

  if (threadIdx.x < 32) {                // wave 0 drives the TDM
    tdm_load_2d(W2, lW2, 64, 64);
    tdm_load_2d(W3, lW3, 64, 64);
    __builtin_amdgcn_s_wait_tensorcnt(0);
  }

  long long er = e0 + col;
  if (er >= N_EDGES) er = N_EDGES - 1;
  float f[EDGE_DIM];
#pragma unroll
  for (int i = 0; i < EDGE_DIM; ++i) f[i] = ef[er * EDGE_DIM + i];

  // ---- layer 1: act1 in A-layout (this lane needs cols where bit1(c)==half)
  float a1[32];
#pragma unroll
  for (int s = 0; s < 16; ++s) {
#pragma unroll
    for (int t = 0; t < 2; ++t) {
      const int c = 4 * s + kb + t;
      float acc = b1[c];
#pragma unroll
      for (int i = 0; i < EDGE_DIM; ++i) acc = fmaf(f[i], W1[c * EDGE_DIM + i], acc);
      a1[2 * s + t] = fmaxf(acc, 0.0f);
    }
  }
  __syncthreads();                       // weights visible to all waves

  // ---- layer 2: act2 = relu(act1 @ W2^T + b2)   (B[k][n] = W2[n][k], from LDS)
  v8f c2[4] = {};
#pragma unroll
  for (int s = 0; s < 16; ++s) {
    v2f a; a.x = a1[2 * s]; a.y = a1[2 * s + 1];
#pragma unroll
    for (int t = 0; t < 4; ++t) {
      const int n = 16 * t + col;
      v2f b; b.x = lW2[n * 64 + 4 * s + kb]; b.y = lW2[n * 64 + 4 * s + kb + 1];
      c2[t] = wmma_f32(a, b, c2[t]);
    }
  }
#pragma unroll
  for (int t = 0; t < 4; ++t) {
    const float bias = b2[16 * t + col];
#pragma unroll
    for (int v = 0; v < 8; ++v) c2[t][v] = fmaxf(c2[t][v] + bias, 0.0f);
  }

  // ---- C-layout -> LDS -> A-layout for next GEMM
#pragma unroll
  for (int t = 0; t < 4; ++t)
#pragma unroll
    for (int v = 0; v < 8; ++v)
      conv[wave][v + 8 * half][16 * t + col] = c2[t][v];
  __syncthreads();
  float a2[32];
#pragma unroll
  for (int s = 0; s < 16; ++s) {
    a2[2 * s]     = conv[wave][col][4 * s + kb];
    a2[2 * s + 1] = conv[wave][col][4 * s + kb + 1];
  }

  // ---- layer 3: w = act2 @ W3^T + b3
  v8f c3[4] = {};
#pragma unroll
  for (int s = 0; s < 16; ++s) {
    v2f a; a.x = a2[2 * s]; a.y = a2[2 * s + 1];
#pragma unroll
    for (int t = 0; t < 4; ++t) {
      const int n = 16 * t + col;
      v2f b; b.x = lW3[n * 64 + 4 * s + kb]; b.y = lW3[n * 64 + 4 * s + kb + 1];
      c3[t] = wmma_f32(a, b, c3[t]);
    }
  }
#pragma unroll
  for (int t = 0; t < 4; ++t) {
    const int n = 16 * t + col;
    const float bias = b3[n];
#pragma unroll
    for (int v = 0; v < 8; ++v) {
      const long long erow = e0 + v + 8 * half;
      if (erow < N_EDGES) w_out[erow * 64 + n] = c3[t][v] + bias;
    }
  }
}

// -------------------------------------------------------------------------
// utility kernels
// -------------------------------------------------------------------------
__global__ void zero_kernel(float* __restrict__ p, long long n) {
  long long i = (long long)blockIdx.x * blockDim.x + threadIdx.x;
  const long long stride = (long long)gridDim.x * blockDim.x;
  for (; i < n; i += stride) p[i] = 0.0f;
}

__global__ void deg_kernel(const int* __restrict__ ei, float* __restrict__ deg) {
  const long long e = (long long)blockIdx.x * blockDim.x + threadIdx.x;
  if (e < N_EDGES) atomicAdd(&deg[ei[e]], 1.0f);  // dst = edge_index[0]
}

__global__ void invdeg_kernel(float* __restrict__ d) {
  const int n = blockIdx.x * blockDim.x + threadIdx.x;
  if (n < N_NODES) d[n] = 1.0f / fmaxf(d[n], 1.0f);
}

__global__ void init_kernel(const float* __restrict__ hx_in,
                            float* __restrict__ hx_cur, float* __restrict__ out) {
  const long long i = (long long)blockIdx.x * blockDim.x + threadIdx.x;
  if (i < (long long)N_NODES * 64) {
    const float v = hx_in[i];
    hx_cur[i] = v;
    out[(i >> 6) * 256 + (i & 63)] = v;  // hxs[0] slice
  }
}

// gather msg = hx[src]*w, scatter-add at dst (mean applied later via inv_deg)
__global__ void scatter_kernel(const int* __restrict__ ei,
                               const float* __restrict__ hx,
                               const float* __restrict__ w,
                               float* __restrict__ agg) {
  const long long idx = (long long)blockIdx.x * blockDim.x + threadIdx.x;
  const long long e = idx >> 4;
  if (e >= N_EDGES) return;
  const int fq = (int)(idx & 15) * 4;
  const int d = ei[e];
  const int s = ei[N_EDGES + e];
  __builtin_prefetch(w + (e + 64) * 64 + fq, 0, 0);  // global_prefetch_b8
  const float4 hv = *(const float4*)(hx + (long long)s * 64 + fq);
  const float4 wv = *(const float4*)(w + e * 64 + fq);
  float* ap = agg + (long long)d * 64 + fq;
  atomicAdd(ap + 0, hv.x * wv.x);
  atomicAdd(ap + 1, hv.y * wv.y);
  atomicAdd(ap + 2, hv.z * wv.z);
  atomicAdd(ap + 3, hv.w * wv.w);
}

// -------------------------------------------------------------------------
// GRU node update: one wave per 16-node tile.
//   inp = sigmoid(hx@Wig^T + big) * agg_mean
//   gi  = inorm(inp@Wih^T), gh = inorm(hx@Whh^T)   (inorm over 192 feats/row)
// Weight matrices (112 KB) staged in LDS by the TDM once per block.
// K-outer/tile-inner per GEMM (12 chains); gi/gh kept as separate loops
// to stay within register budget (fusing them spilled).
// -------------------------------------------------------------------------
__device__ __forceinline__ void inorm12(v8f* g) {
#pragma unroll
  for (int v = 0; v < 8; ++v) {
    float s = 0.0f, q = 0.0f;
#pragma unroll
    for (int t = 0; t < 12; ++t) { const float x = g[t][v]; s += x; q += x * x; }
#pragma unroll
    for (int m = 1; m < 16; m <<= 1) {  // reduce across the 16-lane half
      s += __shfl_xor(s, m, 32);
      q += __shfl_xor(q, m, 32);
    }
    const float mu  = s * (1.0f / 192.0f);
    const float var = q * (1.0f / 192.0f) - mu * mu;
    const float r   = rsqrtf(var + EPS);
#pragma unroll
    for (int t = 0; t < 12; ++t) g[t][v] = (g[t][v] - mu) * r;
  }
}

__global__ void gru_node_kernel(const float* __restrict__ hx,
                                const float* __restrict__ agg,
                                const float* __restrict__ invd,
                                const float* __restrict__ Wih, const float* __restrict__ Whh,
                                const float* __restrict__ bih, const float* __restrict__ bhh,
                                const float* __restrict__ Wig, const float* __restrict__ big,
                                float* __restrict__ hx_new,
                                float* __restrict__ out, int rep) {
  __shared__ float hxs[4][16][68];
  __shared__ float tmp[4][16][68];
  extern __shared__ float dynlds[];  // [Wig 4096][Wih 12288][Whh 12288]
  float* lWig = dynlds;
  float* lWih = dynlds + 4096;
  float* lWhh = dynlds + 4096 + 12288;

  const int wave = threadIdx.x >> 5;
  const int lane = threadIdx.x & 31;
  const int col  = lane & 15;
  const int half = lane >> 4;
  const int kb   = half * 2;
  const long long n0 = ((long long)blockIdx.x * 4 + wave) * 16;

  if (threadIdx.x < 32) {            // wave 0 drives the TDM
    tdm_load_2d(Wig, lWig, 64, 64);
    tdm_load_2d(Wih, lWih, 192, 64);
    tdm_load_2d(Whh, lWhh, 192, 64);
    __builtin_amdgcn_s_wait_tensorcnt(0);
  }

  // stage hx tile in LDS (row-clamped)
  for (int r = 0; r < 16; ++r) {
    long long nr = n0 + r;
    if (nr >= N_NODES) nr = N_NODES - 1;
    for (int c = lane; c < 64; c += 32) hxs[wave][r][c] = hx[nr * 64 + c];
  }
  __syncthreads();                   // hx tiles + TDM'd weights visible

  // hx in A-layout
  float ahx[32];
#pragma unroll
  for (int s = 0; s < 16; ++s) {
    ahx[2 * s]     = hxs[wave][col][4 * s + kb];
    ahx[2 * s + 1] = hxs[wave][col][4 * s + kb + 1];
  }

  // ---- ig = hx @ Wig^T ; inp = sigmoid(ig + big) * (agg * inv_deg)
  v8f inpC[4] = {};
#pragma unroll
  for (int s = 0; s < 16; ++s) {
    v2f a; a.x = ahx[2 * s]; a.y = ahx[2 * s + 1];
#pragma unroll
    for (int t = 0; t < 4; ++t) {
      const int n = 16 * t + col;
      v2f b; b.x = lWig[n * 64 + 4 * s + kb]; b.y = lWig[n * 64 + 4 * s + kb + 1];
      inpC[t] = wmma_f32(a, b, inpC[t]);
    }
  }
#pragma unroll
  for (int t = 0; t < 4; ++t) {
    const int n = 16 * t + col;
    const float bias = big[n];
#pragma unroll
    for (int v = 0; v < 8; ++v) {
      long long nr = n0 + v + 8 * half;
      if (nr >= N_NODES) nr = N_NODES - 1;
      inpC[t][v] = sigmoidf_(inpC[t][v] + bias) * agg[nr * 64 + n] * invd[nr];
    }
  }

  // C-layout -> A-layout via LDS
#pragma unroll
  for (int t = 0; t < 4; ++t)
#pragma unroll
    for (int v = 0; v < 8; ++v)
      tmp[wave][v + 8 * half][16 * t + col] = inpC[t][v];
  __syncthreads();
  float ainp[32];
#pragma unroll
  for (int s = 0; s < 16; ++s) {
    ainp[2 * s]     = tmp[wave][col][4 * s + kb];
    ainp[2 * s + 1] = tmp[wave][col][4 * s + kb + 1];
  }

  // ---- gi = inp @ Wih^T (12 tiles, 12 independent WMMA chains)
  v8f gi[12] = {};
#pragma unroll
  for (int s = 0; s < 16; ++s) {
    v2f a; a.x = ainp[2 * s]; a.y = ainp[2 * s + 1];
#pragma unroll
    for (int t = 0; t < 12; ++t) {
      const int n = 16 * t + col;
      v2f b; b.x = lWih[n * 64 + 4 * s + kb]; b.y = lWih[n * 64 + 4 * s + kb + 1];
      gi[t] = wmma_f32(a, b, gi[t]);
    }
  }

  // ---- gh = hx @ Whh^T (12 tiles, 12 independent WMMA chains)
  v8f gh[12] = {};
#pragma unroll
  for (int s = 0; s < 16; ++s) {
    v2f a; a.x = ahx[2 * s]; a.y = ahx[2 * s + 1];
#pragma unroll
    for (int t = 0; t < 12; ++t) {
      const int n = 16 * t + col;
      v2f b; b.x = lWhh[n * 64 + 4 * s + kb]; b.y = lWhh[n * 64 + 4 * s + kb + 1];
      gh[t] = wmma_f32(a, b, gh[t]);
    }
  }
  inorm12(gi);
  inorm12(gh);

  // ---- gates + hx update
#pragma unroll
  for (int t = 0; t < 4; ++t) {
    const int n = 16 * t + col;
#pragma unroll
    for (int v = 0; v < 8; ++v) {
      const float rg = sigmoidf_(gi[t][v]     + bih[n]       + gh[t][v]     + bhh[n]);
      const float ug = sigmoidf_(gi[t + 4][v] + bih[64 + n]  + gh[t + 4][v] + bhh[64 + n]);
      const float ng = tanhf(gi[t + 8][v] + bih[128 + n] +
                             rg * (gh[t + 8][v] + bhh[128 + n]));
      const float hv = hxs[wave][v + 8 * half][n];
      const float hn = ng + ug * (hv - ng);
      const long long nr = n0 + v + 8 * half;
      if (nr < N_NODES) {
        hx_new[nr * 64 + n] = hn;
        out[nr * 256 + 64 * (rep + 1) + n] = hn;
      }
    }
  }
}

// -------------------------------------------------------------------------
extern "C" void kernel_launch(void* const* d_in, const int* in_sizes, int n_in,
                              void* d_out, int out_size, void* d_ws, size_t ws_size,
                              hipStream_t stream) {
  const float* hx_in = (const float*)d_in[0];
  const int*   ei    = (const int*)d_in[1];
  const float* ef    = (const float*)d_in[2];
  const float* fW1 = (const float*)d_in[3];
  const float* fb1 = (const float*)d_in[4];
  const float* fW2 = (const float*)d_in[5];
  const float* fb2 = (const float*)d_in[6];
  const float* fW3 = (const float*)d_in[7];
  const float* fb3 = (const float*)d_in[8];
  const float* Wih = (const float*)d_in[9];
  const float* Whh = (const float*)d_in[10];
  const float* bih = (const float*)d_in[11];
  const float* bhh = (const float*)d_in[12];
  const float* Wig = (const float*)d_in[13];
  const float* big = (const float*)d_in[14];
  float* out = (float*)d_out;

  float* ws  = (float*)d_ws;
  float* w    = ws;                                   // E*64
  float* agg  = w   + (long long)N_EDGES * 64;        // N*64
  float* hxA  = agg + (long long)N_NODES * 64;        // N*64
  float* hxB  = hxA + (long long)N_NODES * 64;        // N*64
  float* invd = hxB + (long long)N_NODES * 64;        // N

  // per-edge weights via WMMA MLP (W2/W3 TDM-staged: 2*16 KB dynamic LDS)
  fnet_kernel<<<(N_EDGES + 127) / 128, 256, 8192 * sizeof(float), stream>>>(
      ef, fW1, fb1, fW2, fb2, fW3, fb3, w);
  // degree -> inverse degree
  zero_kernel<<<256, 256, 0, stream>>>(invd, (long long)N_NODES);
  deg_kernel<<<(N_EDGES + 255) / 256, 256, 0, stream>>>(ei, invd);
  invdeg_kernel<<<(N_NODES + 255) / 256, 256, 0, stream>>>(invd);
  // hx working copy + first output slice
  init_kernel<<<((long long)N_NODES * 64 + 255) / 256, 256, 0, stream>>>(hx_in, hxA, out);

  float* cur = hxA;
  float* nxt = hxB;
  const size_t gruLds = (size_t)(4096 + 12288 + 12288) * sizeof(float);  // 112 KB
  for (int rep = 0; rep < 3; ++rep) {
    zero_kernel<<<2048, 256, 0, stream>>>(agg, (long long)N_NODES * 64);
    scatter_kernel<<<(int)(((long long)N_EDGES * 16 + 255) / 256), 256, 0, stream>>>(
        ei, cur, w, agg);
    gru_node_kernel<<<(N_NODES + 63) / 64, 128, gruLds, stream>>>(
        cur, agg, invd, Wih, Whh, bih, bhh, Wig, big, nxt, out, rep);
    float* t = cur; cur = nxt; nxt = t;
  }
}